// EvalSO2EquivariantGraphAttention_33036888441080
// MI455X (gfx1250) — compile-verified
//
#include <hip/hip_runtime.h>
#include <hip/hip_bf16.h>

typedef float v2f __attribute__((ext_vector_type(2)));
typedef float v8f __attribute__((ext_vector_type(8)));
typedef int   v4i __attribute__((ext_vector_type(4)));

#if __has_builtin(__builtin_amdgcn_global_load_async_to_lds_b128)
#define HAVE_ASYNC_LDS 1
#else
#define HAVE_ASYNC_LDS 0
#endif

__device__ __constant__ int c_MASK[14] = {0,1,2,3,4,5,6,7,8,10,11,12,13,14};
__device__ __constant__ int c_LIDX[16] = {0,1,1,1,2,2,2,2,2,3,3,3,3,3,3,3};

#if HAVE_ASYNC_LDS
// 16-byte global -> LDS async copy (GLOBAL_LOAD_ASYNC_TO_LDS_B128, ASYNCcnt)
__device__ __forceinline__ void async_copy16(const float* g, float* lds) {
    __attribute__((address_space(1))) v4i* gp =
        (__attribute__((address_space(1))) v4i*)const_cast<float*>(g);
    __attribute__((address_space(3))) v4i* lp =
        (__attribute__((address_space(3))) v4i*)lds;
    __builtin_amdgcn_global_load_async_to_lds_b128(gp, lp, 0, 0);
}
#endif

// XOR-swizzle add within 32-lane group: ds_swizzle_b32 (no bounds-check guards)
template <int XMASK>
__device__ __forceinline__ float swz_add(float v) {
    constexpr int offset = (XMASK << 10) | 0x1f;   // xor_mask[14:10], and_mask[4:0]
    int t = __builtin_amdgcn_ds_swizzle(__float_as_int(v), offset);
    return v + __int_as_float(t);
}

__device__ __forceinline__ float blk_reduce128(float v, float* red, int tid) {
    red[tid] = v;
    __syncthreads();
    #pragma unroll
    for (int s = 64; s > 0; s >>= 1) {
        if (tid < s) red[tid] += red[tid + s];
        __syncthreads();
    }
    float r = red[0];
    __syncthreads();
    return r;
}

// ---------------------------------------------------------------------------
// init per-node softmax/accumulator state (ws is never pre-zeroed for us)
// ---------------------------------------------------------------------------
__global__ void k_init(unsigned int* __restrict__ node_max,
                       float* __restrict__ node_den,
                       float* __restrict__ node_accum, int N) {
    int i = blockIdx.x * blockDim.x + threadIdx.x;
    if (i < N) { node_max[i] = 0u; node_den[i] = 0.f; }   // 0 == ordered-key(-inf)
    if (i < N * 16) node_accum[i] = 0.f;
}

// ---------------------------------------------------------------------------
// Main per-edge kernel: async gather -> LDS, radial MLP (overlapped with DMA),
// Wigner rotation via WMMA f32 16x16x4, grid nonlinearity, attention logit.
// One 128-thread block (4 waves) per edge.
// ---------------------------------------------------------------------------
__global__ __launch_bounds__(128) void k_edge_main(
    const float* __restrict__ xsrc, const float* __restrict__ xdst,
    const float* __restrict__ edist, const float* __restrict__ wig,
    const float* __restrict__ tgrid, const float* __restrict__ fgrid,
    const float* __restrict__ rw1, const float* __restrict__ rb1,
    const float* __restrict__ rlg, const float* __restrict__ rlb,
    const float* __restrict__ rw2, const float* __restrict__ rb2,
    const float* __restrict__ alg, const float* __restrict__ alb,
    const float* __restrict__ adot,
    const int* __restrict__ esrc, const int* __restrict__ edst,
    float* __restrict__ after_Dij, float* __restrict__ fij0,
    float* __restrict__ fijL,
    float* __restrict__ ws_alpha, float* __restrict__ ws_xm2,
    unsigned int* __restrict__ node_max)
{
    const int e   = blockIdx.x;
    const int tid = threadIdx.x;

    __shared__ float ed[128];
    __shared__ float hbuf[64];
    __shared__ float wl[512];
    __shared__ float xmsg[16 * 132];   // RAW concat rows; stride 132 vs bank conflicts
    __shared__ float x0s[128];
    __shared__ float rowsum[16];
    __shared__ float gbuf[64];
    __shared__ float red[128];
    __shared__ float xmv[16];

    const int src = esrc[e];
    const int dst = edst[e];

    // ---- stage raw concat(x_src,x_dst) (16x128) into LDS; overlap with MLP ----
#if HAVE_ASYNC_LDS
    #pragma unroll
    for (int q = 0; q < 4; q++) {
        int chunk = tid + 128 * q;                 // 512 chunks of 16B
        int n = chunk >> 5, cpos = (chunk & 31) * 4;
        const float* g = (cpos < 64) ? (xsrc + (size_t)src * 1024 + n * 64 + cpos)
                                     : (xdst + (size_t)dst * 1024 + n * 64 + (cpos - 64));
        async_copy16(g, &xmsg[n * 132 + cpos]);
    }
#else
    #pragma unroll
    for (int q = 0; q < 16; q++) {
        int idx = tid + 128 * q;
        int n = idx >> 7, c = idx & 127;
        xmsg[n * 132 + c] = (c < 64) ? xsrc[(size_t)src * 1024 + n * 64 + c]
                                     : xdst[(size_t)dst * 1024 + n * 64 + (c - 64)];
    }
#endif

    ed[tid] = edist[(size_t)e * 128 + tid];
    if (tid < 16) rowsum[tid] = 0.f;
    __syncthreads();

    // ---- radial layer 1: 128 -> 64, LayerNorm, SiLU ----
    float hv = 0.f;
    if (tid < 64) {
        float acc = rb1[tid];
        #pragma unroll 4
        for (int k = 0; k < 128; k++) acc = fmaf(ed[k], rw1[k * 64 + tid], acc);
        hv = acc;
    }
    float s1 = blk_reduce128(hv, red, tid);
    float s2 = blk_reduce128(hv * hv, red, tid);
    if (tid < 64) {
        float mu  = s1 * (1.f / 64.f);
        float var = s2 * (1.f / 64.f) - mu * mu;
        float x   = (hv - mu) * rsqrtf(var + 1e-5f) * rlg[tid] + rlb[tid];
        hbuf[tid] = x / (1.f + expf(-x));          // silu
    }
    __syncthreads();

    // ---- radial layer 2: 64 -> 512 (w_edge per (l, c2)) ----
    #pragma unroll
    for (int p = 0; p < 4; p++) {
        int o = tid + 128 * p;
        float acc = rb2[o];
        #pragma unroll 4
        for (int j = 0; j < 64; j++) acc = fmaf(hbuf[j], rw2[j * 512 + o], acc);
        wl[o] = acc;
    }

#if HAVE_ASYNC_LDS
#if __has_builtin(__builtin_amdgcn_s_wait_asynccnt)
    __builtin_amdgcn_s_wait_asynccnt(0);
#else
    asm volatile("s_wait_asynccnt 0x0" ::: "memory");
#endif
#endif
    __syncthreads();   // wl + xmsg both visible

    // ---- WMMA: x_rot = W_masked(16x16, rows 14/15 zero) @ (x_msg * w_edge) ----
    const int lane = tid & 31;
    const int wv   = tid >> 5;
    const int r    = lane & 15;       // M (A) / N (B/D) index
    const int hi   = lane >> 4;
    const int koff = hi * 2;

    // A fragments: lanes 0-15 hold K={4k,4k+1}, lanes 16-31 hold K={4k+2,4k+3}
    v2f afrag[4];
    #pragma unroll
    for (int ks = 0; ks < 4; ks++) {
        float ax = 0.f, ay = 0.f;
        if (r < 14) {
            const float* wrow = wig + (size_t)e * 256 + c_MASK[r] * 16 + ks * 4 + koff;
            ax = wrow[0]; ay = wrow[1];
        }
        afrag[ks].x = ax; afrag[ks].y = ay;
    }

    // l-index of each K row touched by this half-wave (registers after unroll)
    int lA[4], lB[4];
    #pragma unroll
    for (int ks = 0; ks < 4; ks++) {
        int k0 = ks * 4 + koff;
        lA[ks] = (k0 > 0) + (k0 > 3) + (k0 > 8);
        lB[ks] = (k0 + 1 > 0) + (k0 + 1 > 3) + (k0 + 1 > 8);
    }

    float rsum[8];
    #pragma unroll
    for (int j = 0; j < 8; j++) rsum[j] = 0.f;

    #pragma unroll
    for (int tt = 0; tt < 2; tt++) {
        const int t   = wv * 2 + tt;           // column tile 0..7 (wave-uniform)
        const int col = t * 16 + r;
        v8f acc = {0.f, 0.f, 0.f, 0.f, 0.f, 0.f, 0.f, 0.f};
        #pragma unroll
        for (int ks = 0; ks < 4; ks++) {
            const int k0 = ks * 4 + koff;
            v2f b;                              // B 4x16: lane=N, VGPR0/1=K pair
            b.x = xmsg[k0 * 132 + col]       * wl[lA[ks] * 128 + col];
            b.y = xmsg[(k0 + 1) * 132 + col] * wl[lB[ks] * 128 + col];
            acc = __builtin_amdgcn_wmma_f32_16x16x4_f32(
                false, afrag[ks], false, b, (short)0, acc, false, false);
        }
        // D layout: VGPR j, lanes 0-15 -> M=j, lanes 16-31 -> M=j+8, N=lane&15
        #pragma unroll
        for (int j = 0; j < 8; j++) {
            const int M = j + 8 * hi;
            float d = acc[j];
            if (M < 14)
                __builtin_nontemporal_store(
                    d, &after_Dij[(size_t)e * (14 * 128) + M * 128 + col]);
            if (M == 0) {
                __builtin_nontemporal_store(d, &fij0[(size_t)e * 128 + col]);
                x0s[col] = d;
            }
            rsum[j] += d;                       // accumulate across both tiles
        }
    }
    // sum each row over the 16 lanes of the half (ds_swizzle xor, guard-free)
    #pragma unroll
    for (int j = 0; j < 8; j++) {
        float s = rsum[j];
        s = swz_add<1>(s);
        s = swz_add<2>(s);
        s = swz_add<4>(s);
        s = swz_add<8>(s);
        const int M = j + 8 * hi;
        if (r == 0 && M < 14) atomicAdd(&rowsum[M], s);
    }
    __syncthreads();

    // ---- xm = row means (fijL) ----
    if (tid < 14) {
        float m = rowsum[tid] * (1.f / 128.f);
        xmv[tid] = m;
        __builtin_nontemporal_store(m, &fijL[(size_t)e * 14 + tid]);
    }
    __syncthreads();

    // ---- grid roundtrip: silu(to_grid @ xm), from_grid^T @ g ----
    if (tid < 64) {
        float acc = 0.f;
        #pragma unroll
        for (int i = 0; i < 14; i++) acc = fmaf(tgrid[tid * 14 + i], xmv[i], acc);
        gbuf[tid] = acc / (1.f + expf(-acc));    // silu
    }
    __syncthreads();
    if (tid < 14) {
        float acc = 0.f;
        #pragma unroll 8
        for (int b = 0; b < 64; b++) acc = fmaf(fgrid[b * 14 + tid], gbuf[b], acc);
        ws_xm2[(size_t)e * 14 + tid] = acc;      // re-read by k_scatter: keep RT
    }

    // ---- attention logit: LN(128) + smooth leaky relu + dot(alpha_dot) ----
    float xv  = x0s[tid];
    float a1  = blk_reduce128(xv, red, tid);
    float a2  = blk_reduce128(xv * xv, red, tid);
    float mu  = a1 * (1.f / 128.f);
    float var = a2 * (1.f / 128.f) - mu * mu;
    float an  = (xv - mu) * rsqrtf(var + 1e-5f) * alg[tid] + alb[tid];
    float sg  = 1.f / (1.f + expf(-an));
    float sl  = 0.6f * an + 0.4f * an * (2.f * sg - 1.f);   // neg=0.2
    float dv  = blk_reduce128(sl * adot[tid], red, tid);
    if (tid == 0) {
        ws_alpha[e] = dv;
        unsigned int u   = __float_as_uint(dv);
        unsigned int key = (u & 0x80000000u) ? ~u : (u | 0x80000000u);
        atomicMax(&node_max[dst], key);          // ordered-float segment max
    }
}

// ---------------------------------------------------------------------------
// Softmax pass 2: exp(alpha - max[dst]) and denominator segment-sum.
// ---------------------------------------------------------------------------
__global__ void k_softmax(const float* __restrict__ ws_alpha,
                          const unsigned int* __restrict__ node_max,
                          const int* __restrict__ edst,
                          float* __restrict__ ws_ex,
                          float* __restrict__ node_den, int E) {
    int e = blockIdx.x * blockDim.x + threadIdx.x;
    if (e >= E) return;
    int d = edst[e];
    unsigned int key = node_max[d];
    unsigned int u   = (key & 0x80000000u) ? (key ^ 0x80000000u) : ~key;
    float mx = __uint_as_float(u);
    float ex = expf(ws_alpha[e] - mx);
    ws_ex[e] = ex;
    atomicAdd(&node_den[d], ex);
}

// ---------------------------------------------------------------------------
// Normalize alpha, emit aij/vij, rotate back (W^T @ v), segment-sum to nodes.
// 16 lanes per edge (coalesced wigner rows).
// ---------------------------------------------------------------------------
__global__ void k_scatter(const float* __restrict__ ws_ex,
                          const float* __restrict__ node_den,
                          const float* __restrict__ ws_xm2,
                          const float* __restrict__ wig,
                          const int* __restrict__ edst,
                          float* __restrict__ aij, float* __restrict__ vij,
                          float* __restrict__ node_accum, int E) {
    int idx = blockIdx.x * blockDim.x + threadIdx.x;
    int e = idx >> 4, n = idx & 15;
    if (e >= E) return;
    int d = edst[e];
    float alpha = ws_ex[e] / (node_den[d] + 1e-16f);
    if (n == 0) __builtin_nontemporal_store(alpha, &aij[e]);
    float acc = 0.f;
    #pragma unroll
    for (int m = 0; m < 14; m++) {
        float v = ws_xm2[(size_t)e * 14 + m] * alpha;
        if (n == m) __builtin_nontemporal_store(v, &vij[(size_t)e * 14 + m]);
        acc = fmaf(wig[(size_t)e * 256 + c_MASK[m] * 16 + n], v, acc);
    }
    atomicAdd(&node_accum[(size_t)d * 16 + n], acc);
}

// ---------------------------------------------------------------------------
// Final projection: out[b,n,o] = node[b,n] * proj_w[L_INDEX[n], o] (+ proj_b).
// ---------------------------------------------------------------------------
__global__ void k_out(const float* __restrict__ node_accum,
                      const float* __restrict__ pw, const float* __restrict__ pb,
                      float* __restrict__ out, int N) {
    int idx = blockIdx.x * blockDim.x + threadIdx.x;
    if (idx >= N * 1024) return;
    int b = idx >> 10, rem = idx & 1023, n = rem >> 6, o = rem & 63;
    float v = node_accum[b * 16 + n] * pw[c_LIDX[n] * 64 + o];
    if (n == 0) v += pb[o];
    __builtin_nontemporal_store(v, &out[idx]);
}

extern "C" void kernel_launch(void* const* d_in, const int* in_sizes, int n_in,
                              void* d_out, int out_size, void* d_ws, size_t ws_size,
                              hipStream_t stream) {
    const float* xsrc  = (const float*)d_in[0];
    const float* xdst  = (const float*)d_in[1];
    const float* edist = (const float*)d_in[2];
    const float* wig   = (const float*)d_in[3];
    const float* tgrid = (const float*)d_in[4];
    const float* fgrid = (const float*)d_in[5];
    const float* rw1   = (const float*)d_in[6];
    const float* rb1   = (const float*)d_in[7];
    const float* rlg   = (const float*)d_in[8];
    const float* rlb   = (const float*)d_in[9];
    const float* rw2   = (const float*)d_in[10];
    const float* rb2   = (const float*)d_in[11];
    const float* alg   = (const float*)d_in[12];
    const float* alb   = (const float*)d_in[13];
    const float* adot  = (const float*)d_in[14];
    const float* pw    = (const float*)d_in[15];
    const float* pb    = (const float*)d_in[16];
    const int*   esrc  = (const int*)d_in[17];
    const int*   edst  = (const int*)d_in[18];

    const int E = in_sizes[2] / 128;        // edge_distance (E,128)
    const int N = in_sizes[0] / (16 * 64);  // x_src_emb (N,16,64)

    // d_out: [out | after_Dij | fij0 | fijL | aij | vij] (flat, return order)
    float* out       = (float*)d_out;
    float* after_Dij = out + (size_t)N * 16 * 64;
    float* fij0      = after_Dij + (size_t)E * 14 * 128;
    float* fijL      = fij0 + (size_t)E * 128;
    float* aij       = fijL + (size_t)E * 14;
    float* vij       = aij + (size_t)E;

    // workspace: 16E floats + N uints + 17N floats  (~5.9 MB)
    float* ws              = (float*)d_ws;
    float* ws_alpha        = ws;                          // E
    float* ws_ex           = ws + (size_t)E;              // E
    float* ws_xm2          = ws + 2 * (size_t)E;          // 14E
    unsigned int* node_max = (unsigned int*)(ws + 16 * (size_t)E);  // N
    float* node_den        = (float*)(node_max + N);      // N
    float* node_accum      = node_den + N;                // 16N

    k_init<<<(N * 16 + 255) / 256, 256, 0, stream>>>(node_max, node_den, node_accum, N);
    k_edge_main<<<E, 128, 0, stream>>>(xsrc, xdst, edist, wig, tgrid, fgrid,
        rw1, rb1, rlg, rlb, rw2, rb2, alg, alb, adot, esrc, edst,
        after_Dij, fij0, fijL, ws_alpha, ws_xm2, node_max);
    k_softmax<<<(E + 255) / 256, 256, 0, stream>>>(ws_alpha, node_max, edst, ws_ex, node_den, E);
    k_scatter<<<(E * 16 + 255) / 256, 256, 0, stream>>>(ws_ex, node_den, ws_xm2, wig, edst,
                                                        aij, vij, node_accum, E);
    k_out<<<(N * 1024 + 255) / 256, 256, 0, stream>>>(node_accum, pw, pb, out, N);
}